// node_prompt_layer_feature_sum_21534966022317
// MI455X (gfx1250) — compile-verified
//
#include <hip/hip_runtime.h>
#include <hip/hip_bf16.h>

// out[v] = sum over edges (u->v) of emb[u],  emb: [N,64] f32, edges: int32 idx.
//
// Strategy: build CSR-by-destination on device (count -> scan -> bin), then a
// wave-per-node gather kernel accumulates rows in registers (no f32 atomics).
// Edge-index lists are streamed into LDS with CDNA5 async-to-LDS loads,
// double-buffered via ASYNCcnt; indices are consumed as int4 LDS broadcasts
// so 4 row-gathers are in flight per DS read.

#define WPB 8  // waves per block in gather kernel (8 * 32 = 256 threads)

__global__ __launch_bounds__(256) void zero_i32_kernel(int* p, int n) {
    int i = blockIdx.x * blockDim.x + threadIdx.x;
    if (i < n) p[i] = 0;
}

__global__ __launch_bounds__(256) void zero_f32_kernel(float* p, int n) {
    int i = blockIdx.x * blockDim.x + threadIdx.x;
    if (i < n) p[i] = 0.0f;
}

// Count in-degree of each node.
__global__ __launch_bounds__(256) void count_kernel(const int* __restrict__ dst,
                                                    int* cnt, int n_edges) {
    int e = blockIdx.x * blockDim.x + threadIdx.x;
    if (e < n_edges) atomicAdd(&cnt[dst[e]], 1);
}

// Single-workgroup exclusive scan: offsets[i] = sum(cnt[0..i-1]); offsets[n] = total.
// Also initializes cursor[i] = offsets[i] (cnt and cursor may alias).
__global__ __launch_bounds__(1024) void scan_kernel(const int* cnt, int* offsets,
                                                    int* cursor, int n) {
    __shared__ int sdata[1024];
    __shared__ int s_carry;
    const int t = threadIdx.x;
    if (t == 0) s_carry = 0;
    __syncthreads();
    for (int base = 0; base < n; base += 1024) {
        const int i = base + t;
        const int x = (i < n) ? cnt[i] : 0;
        sdata[t] = x;
        __syncthreads();
        // Hillis-Steele inclusive scan
        for (int off = 1; off < 1024; off <<= 1) {
            int v = (t >= off) ? sdata[t - off] : 0;
            __syncthreads();
            sdata[t] += v;
            __syncthreads();
        }
        const int incl = sdata[t];
        const int excl = incl - x;
        const int carry = s_carry;
        if (i < n) {
            offsets[i] = carry + excl;
            cursor[i]  = carry + excl;
        }
        __syncthreads();
        if (t == 1023) s_carry = carry + incl;
        __syncthreads();
    }
    if (t == 0) offsets[n] = s_carry;
}

// Bin source indices by destination.
__global__ __launch_bounds__(256) void bin_kernel(const int* __restrict__ src,
                                                  const int* __restrict__ dst,
                                                  int* cursor, int* __restrict__ esrc,
                                                  int n_edges) {
    int e = blockIdx.x * blockDim.x + threadIdx.x;
    if (e < n_edges) {
        int pos = atomicAdd(&cursor[dst[e]], 1);
        esrc[pos] = src[e];
    }
}

// One wave32 per node: stream edge-index list into LDS with async-to-LDS loads
// (double-buffered on ASYNCcnt), gather embedding rows, accumulate in registers,
// write the 64-float result once. No float atomics anywhere.
__global__ __launch_bounds__(256) void gather_kernel(const float* __restrict__ emb,
                                                     const int* __restrict__ offsets,
                                                     const int* __restrict__ esrc,
                                                     float* __restrict__ out,
                                                     int n_nodes, int n_edges) {
    __shared__ int ebuf[WPB][2][32];
    const int lane = threadIdx.x & 31;
    const int wv   = threadIdx.x >> 5;
    const int node = blockIdx.x * WPB + wv;
    if (node >= n_nodes) return;  // uniform per wave

    const int begin = offsets[node];
    const int end   = offsets[node + 1];
    const int total = end - begin;

    float2 acc = make_float2(0.0f, 0.0f);
    const float* embp = emb + 2 * lane;  // this lane owns features [2*lane, 2*lane+1]

    if (total > 0) {
        const int nIter = (total + 31) >> 5;
        // Per-lane LDS byte addresses of this wave's two 32-entry buffers.
        // (Truncating a generic LDS pointer yields the LDS byte offset: the
        //  shared aperture lives entirely in ADDR[63:32].)
        const unsigned lds0 = (unsigned)(size_t)(&ebuf[wv][0][lane]);
        const unsigned lds1 = (unsigned)(size_t)(&ebuf[wv][1][lane]);

        // Prefetch iteration 0 into buffer 0 (GVS mode: SGPR base + VGPR offset).
        // Tail lanes clamp to a valid esrc entry, so all 32 slots always hold
        // legal node indices (garbage entries are simply not accumulated).
        {
            int idx = begin + lane;
            if (idx >= n_edges) idx = n_edges - 1;
            unsigned voff = (unsigned)idx << 2;
            asm volatile("global_load_async_to_lds_b32 %0, %1, %2"
                         :: "v"(lds0), "v"(voff), "s"(esrc) : "memory");
        }

        for (int it = 0; it < nIter; ++it) {
            if (it + 1 < nIter) {
                // Issue prefetch of next chunk, then wait until only it remains
                // outstanding -> current chunk's buffer is complete in LDS.
                int idx = begin + (it + 1) * 32 + lane;
                if (idx >= n_edges) idx = n_edges - 1;
                unsigned voff = (unsigned)idx << 2;
                unsigned ldsn = ((it + 1) & 1) ? lds1 : lds0;
                asm volatile("global_load_async_to_lds_b32 %0, %1, %2"
                             :: "v"(ldsn), "v"(voff), "s"(esrc) : "memory");
                asm volatile("s_wait_asynccnt 0x1" ::: "memory");
            } else {
                asm volatile("s_wait_asynccnt 0x0" ::: "memory");
            }

            const int* buf = &ebuf[wv][it & 1][0];
            int cnt = total - it * 32;
            if (cnt >= 32) {
                // Fast path: full chunk. int4 LDS broadcast (ds_load_b128),
                // 4 unconditional row-gathers per index quad -> high MLP.
#pragma unroll
                for (int q = 0; q < 8; ++q) {
                    const int4 s4 = ((const int4*)buf)[q];
                    const float2 va = *(const float2*)(embp + (size_t)s4.x * 64);
                    const float2 vb = *(const float2*)(embp + (size_t)s4.y * 64);
                    const float2 vc = *(const float2*)(embp + (size_t)s4.z * 64);
                    const float2 vd = *(const float2*)(embp + (size_t)s4.w * 64);
                    acc.x += (va.x + vb.x) + (vc.x + vd.x);
                    acc.y += (va.y + vb.y) + (vc.y + vd.y);
                }
            } else {
                // Tail chunk: guarded scalar loop.
                for (int j = 0; j < cnt; ++j) {
                    const int s = buf[j];
                    const float2 v = *(const float2*)(embp + (size_t)s * 64);
                    acc.x += v.x;
                    acc.y += v.y;
                }
            }
        }
    }

    *(float2*)(out + (size_t)node * 64 + 2 * lane) = acc;
}

// ---- Fallback path (workspace too small): atomic scatter --------------------
__global__ __launch_bounds__(256) void scatter_atomic_kernel(const float* __restrict__ emb,
                                                             const int* __restrict__ src,
                                                             const int* __restrict__ dst,
                                                             float* out, int n_edges) {
    int tid = blockIdx.x * blockDim.x + threadIdx.x;
    int e = tid >> 4;   // 16 threads per edge
    int c = tid & 15;   // 4 floats per thread
    if (e >= n_edges) return;
    const int s = src[e];
    const int d = dst[e];
    const float4 v = *(const float4*)(emb + (size_t)s * 64 + c * 4);
    float* op = out + (size_t)d * 64 + c * 4;
    atomicAdd(op + 0, v.x);
    atomicAdd(op + 1, v.y);
    atomicAdd(op + 2, v.z);
    atomicAdd(op + 3, v.w);
}

extern "C" void kernel_launch(void* const* d_in, const int* in_sizes, int n_in,
                              void* d_out, int out_size, void* d_ws, size_t ws_size,
                              hipStream_t stream) {
    const float* emb = (const float*)d_in[0];
    const int*   src = (const int*)d_in[1];
    const int*   dst = (const int*)d_in[2];
    float* out = (float*)d_out;

    const int n_edges = in_sizes[1];
    const int n_nodes = in_sizes[0] / 64;

    const size_t needed = ((size_t)n_edges + 2 * (size_t)n_nodes + 2) * sizeof(int);

    if (ws_size >= needed) {
        // ws layout: esrc[E] | offsets[N+1] | cursor[N]
        int* esrc    = (int*)d_ws;
        int* offsets = esrc + n_edges;
        int* cursor  = offsets + (n_nodes + 1);

        zero_i32_kernel<<<(n_nodes + 255) / 256, 256, 0, stream>>>(cursor, n_nodes);
        count_kernel<<<(n_edges + 255) / 256, 256, 0, stream>>>(dst, cursor, n_edges);
        scan_kernel<<<1, 1024, 0, stream>>>(cursor, offsets, cursor, n_nodes);
        bin_kernel<<<(n_edges + 255) / 256, 256, 0, stream>>>(src, dst, cursor, esrc, n_edges);
        gather_kernel<<<(n_nodes + WPB - 1) / WPB, 256, 0, stream>>>(
            emb, offsets, esrc, out, n_nodes, n_edges);
    } else {
        // Fallback: zero output, then per-edge atomic scatter.
        const int total = n_nodes * 64;
        zero_f32_kernel<<<(total + 255) / 256, 256, 0, stream>>>(out, total);
        const long long threads = (long long)n_edges * 16;
        scatter_atomic_kernel<<<(int)((threads + 255) / 256), 256, 0, stream>>>(
            emb, src, dst, out, n_edges);
    }
}